// GCADecoder_7533372637721
// MI455X (gfx1250) — compile-verified
//
#include <hip/hip_runtime.h>

typedef float v2f __attribute__((ext_vector_type(2)));
typedef float v8f __attribute__((ext_vector_type(8)));

#define H2 256
#define H1 128
#define H0 64
#define OC 8

// ---------- degree / normalization ----------
__global__ void init_val_kernel(float* __restrict__ p, float v, int n) {
  int i = blockIdx.x * blockDim.x + threadIdx.x;
  if (i < n) p[i] = v;
}

__global__ void deg_accum_kernel(const int* __restrict__ dst, int E, float* __restrict__ deg) {
  int i = blockIdx.x * blockDim.x + threadIdx.x;
  if (i < E) atomicAdd(&deg[dst[i]], 1.0f);
}

__global__ void to_rsqrt_kernel(float* __restrict__ p, int n) {
  int i = blockIdx.x * blockDim.x + threadIdx.x;
  if (i < n) p[i] = rsqrtf(p[i]);   // deg >= 1 always (self loop)
}

// ---------- dense GEMM via V_WMMA_F32_16X16X4_F32, A staged in LDS ----------
// Block: 256 threads = 8 waves covering a (16*WM) x (16*WN) macro-tile, WM*WN==8.
// A panel (ROWS x K) loaded cooperatively (coalesced b128) into padded LDS;
// per K-step each lane reads its two A values with one conflict-free ds_load_b64.
// Out-of-range rows/cols are CLAMPED, not masked: they only feed output
// elements we never store, so no exec-mask juggling and no zero-fill needed.
template <int K, int NC, int WM>
__global__ __launch_bounds__(256)
void wmma_gemm_f32_lds_kernel(const float* __restrict__ X, const float* __restrict__ W,
                              float* __restrict__ G, int D) {
  constexpr int WN   = 8 / WM;
  constexpr int ROWS = 16 * WM;                 // block rows (M)
  constexpr int BN   = 16 * WN;                 // block cols (N)
  constexpr int LDK  = K + 4;                   // pad => bank-conflict-free b64
  constexpr int NBLK = (NC + BN - 1) / BN;
  __shared__ float As[ROWS * LDK];

  const int tid     = threadIdx.x;
  const int bm      = (int)blockIdx.x / NBLK;
  const int bn      = (int)blockIdx.x - bm * NBLK;
  const int rowBase = bm * ROWS;

  constexpr int F4 = K / 4;                     // float4 per A row
#pragma unroll
  for (int f = tid; f < ROWS * F4; f += 256) {
    const int r  = f / F4;
    const int c4 = f - r * F4;
    int gr = rowBase + r;
    gr = gr < D ? gr : D - 1;                   // clamp (safe, see above)
    const float4 v = ((const float4*)(X + (size_t)gr * K))[c4];
    float* dp = &As[r * LDK + c4 * 4];
    dp[0] = v.x; dp[1] = v.y; dp[2] = v.z; dp[3] = v.w;
  }
  __syncthreads();

  const int wave = tid >> 5;
  const int lane = tid & 31;
  const int wm   = wave / WN;
  const int wn   = wave - wm * WN;
  const int half = lane >> 4;                   // 0: K pair {k,k+1}; 1: {k+2,k+3}
  const int l    = lane & 15;

  const int col  = bn * BN + wn * 16 + l;       // B/C column for this lane
  const int colC = (col < NC) ? col : (NC - 1); // clamp for load safety
  const float* __restrict__ arow = &As[(wm * 16 + l) * LDK];
  const float* __restrict__ wcol = W + colC;

  v8f acc = {};
#pragma unroll
  for (int k = 0; k < K; k += 4) {
    const int ka = k + half * 2;
    const v2f a = *(const v2f*)(arow + ka);     // ds_load_b64
    v2f b;
    b.x = wcol[ka * NC];                        // NC is constexpr => shifts
    b.y = wcol[(ka + 1) * NC];
    acc = __builtin_amdgcn_wmma_f32_16x16x4_f32(false, a, false, b,
                                                (short)0, acc, false, false);
  }

#pragma unroll
  for (int v = 0; v < 8; ++v) {
    const int r = rowBase + wm * 16 + half * 8 + v;  // C/D: VGPR v -> M=v (+8 upper)
    if (r < D && col < NC) G[(size_t)r * NC + col] = acc[v];
  }
}

// ---------- edge scatter: out[dst] += g[src>>SHIFT] * dinv[src]*dinv[dst] ----------
template <int F, int TPE, int SHIFT>
__global__ __launch_bounds__(256)
void scatter_edges_kernel(const int* __restrict__ ei, int E,
                          const float* __restrict__ g, const float* __restrict__ dinv,
                          float* __restrict__ out) {
  const int t = blockIdx.x * blockDim.x + threadIdx.x;
  const int e = t / TPE;
  const int f0 = t - e * TPE;
  if (e >= E) return;
  const int s = ei[e];
  const int d = ei[E + e];
  const float norm = dinv[s] * dinv[d];
  const float* gr = g + (size_t)(s >> SHIFT) * F;
  float* orow = out + (size_t)d * F;
#pragma unroll
  for (int f = f0; f < F; f += TPE)
    atomicAdd(&orow[f], gr[f] * norm);          // non-returning global_atomic_add_f32
}

// ---------- epilogue: self-loop term + bias (+ReLU), in place ----------
template <int F, int SHIFT, bool RELU>
__global__ __launch_bounds__(256)
void epilogue_kernel(float* __restrict__ agg, const float* __restrict__ g,
                     const float* __restrict__ dinv, const float* __restrict__ bias,
                     int Nn) {
  const int idx = blockIdx.x * blockDim.x + threadIdx.x;
  if (idx >= Nn * F) return;
  const int i = idx / F;
  const int f = idx - i * F;
  const float di = dinv[i];
  float v = agg[idx] + g[(size_t)(i >> SHIFT) * F + f] * (di * di) + bias[f];
  if (RELU) v = fmaxf(v, 0.0f);
  agg[idx] = v;
}

static inline int cdiv_ll(long long a, long long b) { return (int)((a + b - 1) / b); }

extern "C" void kernel_launch(void* const* d_in, const int* in_sizes, int n_in,
                              void* d_out, int out_size, void* d_ws, size_t ws_size,
                              hipStream_t stream) {
  const float* z  = (const float*)d_in[0];
  const int*   e1 = (const int*)d_in[1];
  const int*   e2 = (const int*)d_in[2];
  const int*   e3 = (const int*)d_in[3];
  const int*   e4 = (const int*)d_in[4];
  const float* W1 = (const float*)d_in[5];
  const float* b1 = (const float*)d_in[6];
  const float* W2 = (const float*)d_in[7];
  const float* b2 = (const float*)d_in[8];
  const float* W3 = (const float*)d_in[9];
  const float* b3 = (const float*)d_in[10];
  const float* W4 = (const float*)d_in[11];
  const float* b4 = (const float*)d_in[12];

  const int N  = in_sizes[0] / H2;   // 25000
  const int E1 = in_sizes[1] / 2;
  const int E2 = in_sizes[2] / 2;
  const int E3 = in_sizes[3] / 2;
  const int E4 = in_sizes[4] / 2;

  float* ws    = (float*)d_ws;
  float* dinv1 = ws;                 // N
  float* dinv2 = dinv1 + N;          // 2N
  float* dinv3 = dinv2 + 2 * N;      // 4N
  float* dinv4 = dinv3 + 4 * N;      // 8N
  float* slotA = dinv4 + 8 * N;      // max(N*256, 2N*128, 4N*64) floats
  float* slotG = slotA + (size_t)N * H2;

  const int T = 256;

  // degrees (self-loop => init 1.0) -> dinv, 15N counters contiguous
  init_val_kernel<<<cdiv_ll(15LL * N, T), T, 0, stream>>>(dinv1, 1.0f, 15 * N);
  deg_accum_kernel<<<cdiv_ll(E1, T), T, 0, stream>>>(e1 + E1, E1, dinv1);
  deg_accum_kernel<<<cdiv_ll(E2, T), T, 0, stream>>>(e2 + E2, E2, dinv2);
  deg_accum_kernel<<<cdiv_ll(E3, T), T, 0, stream>>>(e3 + E3, E3, dinv3);
  deg_accum_kernel<<<cdiv_ll(E4, T), T, 0, stream>>>(e4 + E4, E4, dinv4);
  to_rsqrt_kernel<<<cdiv_ll(15LL * N, T), T, 0, stream>>>(dinv1, 15 * N);

  // ---- layer 1: g1 = z @ W1 [N x 256]; aggregate over N nodes ----
  {
    const int blocks = cdiv_ll(N, 16) * 2;   // NBLK = 256/128 = 2
    wmma_gemm_f32_lds_kernel<256, 256, 1><<<blocks, T, 0, stream>>>(z, W1, slotG, N);
    hipMemsetAsync(slotA, 0, (size_t)N * H2 * sizeof(float), stream);
    scatter_edges_kernel<H2, 32, 0><<<cdiv_ll((long long)E1 * 32, T), T, 0, stream>>>(
        e1, E1, slotG, dinv1, slotA);
    epilogue_kernel<H2, 0, true><<<cdiv_ll((long long)N * H2, T), T, 0, stream>>>(
        slotA, slotG, dinv1, b1, N);
  }

  // ---- layer 2: g2 = relu1 @ W2 [N x 128]; nodes 2N, h[j] = g2[j>>1] ----
  {
    const int blocks = cdiv_ll(N, 16);       // NBLK = 1
    wmma_gemm_f32_lds_kernel<256, 128, 1><<<blocks, T, 0, stream>>>(slotA, W2, slotG, N);
    hipMemsetAsync(slotA, 0, (size_t)(2 * N) * H1 * sizeof(float), stream);
    scatter_edges_kernel<H1, 32, 1><<<cdiv_ll((long long)E2 * 32, T), T, 0, stream>>>(
        e2, E2, slotG, dinv2, slotA);
    epilogue_kernel<H1, 1, true><<<cdiv_ll((long long)(2 * N) * H1, T), T, 0, stream>>>(
        slotA, slotG, dinv2, b2, 2 * N);
  }

  // ---- layer 3: g3 = relu2 @ W3 [2N x 64]; nodes 4N ----
  {
    const int blocks = cdiv_ll(2LL * N, 32); // ROWS=32, NBLK=1
    wmma_gemm_f32_lds_kernel<128, 64, 2><<<blocks, T, 0, stream>>>(slotA, W3, slotG, 2 * N);
    hipMemsetAsync(slotA, 0, (size_t)(4 * N) * H0 * sizeof(float), stream);
    scatter_edges_kernel<H0, 32, 1><<<cdiv_ll((long long)E3 * 32, T), T, 0, stream>>>(
        e3, E3, slotG, dinv3, slotA);
    epilogue_kernel<H0, 1, true><<<cdiv_ll((long long)(4 * N) * H0, T), T, 0, stream>>>(
        slotA, slotG, dinv3, b3, 4 * N);
  }

  // ---- layer 4: g4 = relu3 @ W4 [4N x 8]; nodes 8N, result -> d_out (no relu) ----
  {
    float* out = (float*)d_out;
    const int blocks = cdiv_ll(4LL * N, 128); // ROWS=128, NBLK=1
    wmma_gemm_f32_lds_kernel<64, 8, 8><<<blocks, T, 0, stream>>>(slotA, W4, slotG, 4 * N);
    hipMemsetAsync(out, 0, (size_t)(8 * N) * OC * sizeof(float), stream);
    scatter_edges_kernel<OC, 8, 1><<<cdiv_ll((long long)E4 * 8, T), T, 0, stream>>>(
        e4, E4, slotG, dinv4, out);
    epilogue_kernel<OC, 1, false><<<cdiv_ll((long long)(8 * N) * OC, T), T, 0, stream>>>(
        out, slotG, dinv4, b4, 8 * N);
  }
}